// NodeModel_5188320494485
// MI455X (gfx1250) — compile-verified
//
#include <hip/hip_runtime.h>
#include <hip/hip_bf16.h>

typedef float v2f __attribute__((ext_vector_type(2)));
typedef float v8f __attribute__((ext_vector_type(8)));

#define D_IN   144   // f_x(64) + f_e(16) + f_u(64)
#define HDIM   128
#define FOUT   64
#define LDS_STRIDE 162   // even (8B-aligned rows), 162%64=34 -> conflict-free for 16 rows
#define WAVES_PER_BLOCK 4

// ---------------- kernel 0: zero the scatter workspace ----------------
__global__ void zero_ws_kernel(float* __restrict__ ws, long long n) {
    long long i = (long long)blockIdx.x * blockDim.x + threadIdx.x;
    long long stride = (long long)gridDim.x * blockDim.x;
    for (; i < n; i += stride) ws[i] = 0.0f;
}

// ---------------- kernel 1: scatter-add edge features + counts ----------------
__global__ void scatter_kernel(const float* __restrict__ edge_attr,
                               const int*   __restrict__ col,   // edge_index row 1
                               float* __restrict__ sums,        // [n_nodes,16]
                               float* __restrict__ cnt,         // [n_nodes]
                               long long n_edges) {
    long long t = (long long)blockIdx.x * blockDim.x + threadIdx.x;
    long long total = n_edges * 16;
    if (t >= total) return;
    long long e = t >> 4;
    int f = (int)(t & 15);
    int c = col[e];
    atomicAdd(&sums[(size_t)c * 16 + f], edge_attr[e * 16 + f]);
    if (f == 0) atomicAdd(&cnt[c], 1.0f);
}

// ---------------- kernel 2: fused concat + 2-layer MLP via fp32 WMMA ----------------
__global__ __launch_bounds__(32 * WAVES_PER_BLOCK)
void mlp_wmma_kernel(const float* __restrict__ x,      // [N,64]
                     const float* __restrict__ sums,   // [N,16]
                     const float* __restrict__ cnt,    // [N]
                     const float* __restrict__ u,      // [64,64]
                     const int*   __restrict__ batch,  // [N]
                     const float* __restrict__ W1,     // [144,128]
                     const float* __restrict__ b1,     // [128]
                     const float* __restrict__ W2,     // [128,64]
                     const float* __restrict__ b2,     // [64]
                     float* __restrict__ out,          // [N,64]
                     int n_nodes) {
    __shared__ float lds[WAVES_PER_BLOCK][16 * LDS_STRIDE];

    const int lane = threadIdx.x & 31;
    const int wave = threadIdx.x >> 5;
    const int tile = blockIdx.x * WAVES_PER_BLOCK + wave;
    const int row0 = tile * 16;
    float* t = lds[wave];

    // ---- stage 16x144 input tile: [x | e_agg | u[batch]] ----
    for (int idx = lane; idx < 16 * D_IN; idx += 32) {
        int r = idx / D_IN;
        int c = idx - r * D_IN;
        int node = row0 + r;
        float v = 0.0f;
        if (node < n_nodes) {
            if (c < 64) {
                v = x[(size_t)node * 64 + c];
            } else if (c < 80) {
                float s = sums[(size_t)node * 16 + (c - 64)];
                float n = cnt[node];
                v = s / fmaxf(n, 1.0f);
            } else {
                v = u[(size_t)batch[node] * 64 + (c - 80)];
            }
        }
        t[r * LDS_STRIDE + c] = v;
    }

    const int half = lane >> 4;       // 0: lanes 0-15, 1: lanes 16-31
    const int nlo  = lane & 15;

    // ---- GEMM1: [16,144] x [144,128] -> [16,128] ----
    v8f acc[8];
#pragma unroll
    for (int i = 0; i < 8; i++)
#pragma unroll
        for (int j = 0; j < 8; j++) acc[i][j] = 0.0f;

    for (int k0 = 0; k0 < D_IN; k0 += 4) {
        int ka = k0 + 2 * half;       // A layout: lanes 0-15 hold K=k0,k0+1; 16-31 hold K=k0+2,k0+3
        v2f a;
        a.x = t[nlo * LDS_STRIDE + ka];
        a.y = t[nlo * LDS_STRIDE + ka + 1];
#pragma unroll
        for (int nt = 0; nt < 8; nt++) {
            int ncol = nt * 16 + nlo;
            v2f b;
            b.x = W1[(size_t)ka * HDIM + ncol];
            b.y = W1[(size_t)(ka + 1) * HDIM + ncol];
            acc[nt] = __builtin_amdgcn_wmma_f32_16x16x4_f32(
                false, a, false, b, (short)0, acc[nt], false, false);
        }
    }

    // ---- bias + ReLU; transpose hidden back to LDS (reuse tile buffer) ----
#pragma unroll
    for (int nt = 0; nt < 8; nt++) {
        int ncol = nt * 16 + nlo;
        float bb = b1[ncol];
#pragma unroll
        for (int v = 0; v < 8; v++) {
            float h = fmaxf(acc[nt][v] + bb, 0.0f);
            // C layout: row = v + 8*half, col = ncol
            t[(v + 8 * half) * LDS_STRIDE + ncol] = h;
        }
    }

    // ---- GEMM2: [16,128] x [128,64] -> [16,64] ----
    v8f acc2[4];
#pragma unroll
    for (int i = 0; i < 4; i++)
#pragma unroll
        for (int j = 0; j < 8; j++) acc2[i][j] = 0.0f;

    for (int k0 = 0; k0 < HDIM; k0 += 4) {
        int ka = k0 + 2 * half;
        v2f a;
        a.x = t[nlo * LDS_STRIDE + ka];
        a.y = t[nlo * LDS_STRIDE + ka + 1];
#pragma unroll
        for (int nt = 0; nt < 4; nt++) {
            int ncol = nt * 16 + nlo;
            v2f b;
            b.x = W2[(size_t)ka * FOUT + ncol];
            b.y = W2[(size_t)(ka + 1) * FOUT + ncol];
            acc2[nt] = __builtin_amdgcn_wmma_f32_16x16x4_f32(
                false, a, false, b, (short)0, acc2[nt], false, false);
        }
    }

    // ---- bias + store ----
#pragma unroll
    for (int nt = 0; nt < 4; nt++) {
        int ncol = nt * 16 + nlo;
        float bb = b2[ncol];
#pragma unroll
        for (int v = 0; v < 8; v++) {
            int node = row0 + v + 8 * half;
            if (node < n_nodes)
                out[(size_t)node * FOUT + ncol] = acc2[nt][v] + bb;
        }
    }
}

extern "C" void kernel_launch(void* const* d_in, const int* in_sizes, int n_in,
                              void* d_out, int out_size, void* d_ws, size_t ws_size,
                              hipStream_t stream) {
    const float* x         = (const float*)d_in[0];
    const float* edge_attr = (const float*)d_in[1];
    const float* u         = (const float*)d_in[2];
    const float* W1        = (const float*)d_in[3];
    const float* b1        = (const float*)d_in[4];
    const float* W2        = (const float*)d_in[5];
    const float* b2        = (const float*)d_in[6];
    const int*   edge_index = (const int*)d_in[7];
    const int*   batch      = (const int*)d_in[8];
    float* out = (float*)d_out;

    const long long n_nodes = in_sizes[0] / 64;   // F_X = 64
    const long long n_edges = in_sizes[1] / 16;   // F_E = 16

    float* sums = (float*)d_ws;                      // [n_nodes,16]
    float* cnt  = sums + (size_t)n_nodes * 16;       // [n_nodes]

    // 0) zero scatter workspace
    {
        long long n = n_nodes * 17;
        int blocks = (int)((n + 255) / 256);
        if (blocks > 65535) blocks = 65535;
        zero_ws_kernel<<<blocks, 256, 0, stream>>>((float*)d_ws, n);
    }

    // 1) scatter-add edges (col = edge_index row 1)
    {
        long long total = n_edges * 16;
        unsigned blocks = (unsigned)((total + 255) / 256);
        scatter_kernel<<<blocks, 256, 0, stream>>>(edge_attr, edge_index + n_edges,
                                                   sums, cnt, n_edges);
    }

    // 2) fused MLP with WMMA
    {
        long long tiles  = (n_nodes + 15) / 16;
        unsigned  blocks = (unsigned)((tiles + WAVES_PER_BLOCK - 1) / WAVES_PER_BLOCK);
        mlp_wmma_kernel<<<blocks, 32 * WAVES_PER_BLOCK, 0, stream>>>(
            x, sums, cnt, u, batch, W1, b1, W2, b2, out, (int)n_nodes);
    }
}